// TransLayer_644245095163
// MI455X (gfx1250) — compile-verified
//
#include <hip/hip_runtime.h>
#include <hip/hip_bf16.h>

// ---------------------------------------------------------------------------
// Nystromformer layer for MI455X (gfx1250, wave32).
// HBM-bound on the 256MB attn output -> FP32 WMMA (V_WMMA_F32_16X16X4_F32)
// for every GEMM: exact reference numerics; matrix FLOPs (~64 GF) sit far
// below the ~20us HBM floor, so lower precision buys nothing here.
// GEMM: each wave owns a 16x64 C tile (4 accumulators, shared A fragment,
// 4 independent WMMA chains); 128-thread block = 64x64 macro-tile.
// K-loop is software-pipelined (double-buffered fragments) so the five
// global_load_b64 of step i+1 are in flight while step i's WMMAs execute --
// avoids the s_wait_loadcnt 0x0 full-stall before every WMMA.
// ---------------------------------------------------------------------------

#define DIMC    512
#define HEADS   8
#define DHEAD   64
#define MLAND   256
#define NSEQ    2048
#define BATCHB  2
#define KCONV   33
#define PINV_ITERS 6
#define BH      (BATCHB * HEADS)     // 16
#define ROWS    (BATCHB * NSEQ)      // 4096

static constexpr float QSCALE = 0.125f;   // DHEAD^-0.5

typedef __attribute__((ext_vector_type(2))) float v2f;
typedef __attribute__((ext_vector_type(8))) float v8f;

// ---------------------------------------------------------------------------
// Batched GEMM: C[z] = alpha * A[z] @ op(B[z])
// A: MxK row-major (lda). B: KxN (BT=0, ldb) or row-major NxK (BT=1, ldb).
// Fragment layouts per CDNA5 ISA 7.12.2:
//   A 16x4 f32: lanes 0-15 hold (M=lane, K=k0,k0+1); lanes 16-31 K=k0+2,k0+3
//   C/D 16x16 : VGPR i -> M=i (lanes 0-15) / M=8+i (lanes 16-31), N=lane%16
// ---------------------------------------------------------------------------
template<int BT>
__global__ __launch_bounds__(128)
void gemm_f32_wmma(const float* __restrict__ A, const float* __restrict__ B,
                   float* __restrict__ C,
                   int M, int N, int K, int lda, int ldb, int ldc,
                   long long sA, long long sB, long long sC, float alpha)
{
    const int z = blockIdx.z;
    A += (long long)z * sA;
    B += (long long)z * sB;
    C += (long long)z * sC;

    const int lane = threadIdx.x & 31;
    const int wave = threadIdx.x >> 5;
    const int row0 = blockIdx.y * 64 + wave * 16;   // wave-uniform
    const int col0 = blockIdx.x * 64;
    if (row0 >= M || col0 >= N) return;             // wave-uniform: EXEC all-1s

    const int half = lane >> 4;                     // which K pair this lane holds
    const int l16  = lane & 15;

    const float* Ap = A + (long long)(row0 + l16) * lda + half * 2;
    const long long ldbl = ldb;

    v8f acc0 = {}, acc1 = {}, acc2 = {}, acc3 = {};

    if (BT) {
        const float* Bp0 = B + (long long)(col0      + l16) * ldbl + half * 2;
        const float* Bp1 = B + (long long)(col0 + 16 + l16) * ldbl + half * 2;
        const float* Bp2 = B + (long long)(col0 + 32 + l16) * ldbl + half * 2;
        const float* Bp3 = B + (long long)(col0 + 48 + l16) * ldbl + half * 2;
        // pipeline prologue: issue step-0 loads
        v2f a  = *(const v2f*)Ap;
        v2f b0 = *(const v2f*)Bp0;
        v2f b1 = *(const v2f*)Bp1;
        v2f b2 = *(const v2f*)Bp2;
        v2f b3 = *(const v2f*)Bp3;
        for (int k0 = 4; k0 < K; k0 += 4) {
            Ap += 4; Bp0 += 4; Bp1 += 4; Bp2 += 4; Bp3 += 4;
            // issue step-(i+1) loads into distinct regs, then compute step i
            v2f an  = *(const v2f*)Ap;
            v2f b0n = *(const v2f*)Bp0;
            v2f b1n = *(const v2f*)Bp1;
            v2f b2n = *(const v2f*)Bp2;
            v2f b3n = *(const v2f*)Bp3;
            acc0 = __builtin_amdgcn_wmma_f32_16x16x4_f32(false, a, false, b0, (short)0, acc0, false, false);
            acc1 = __builtin_amdgcn_wmma_f32_16x16x4_f32(false, a, false, b1, (short)0, acc1, false, false);
            acc2 = __builtin_amdgcn_wmma_f32_16x16x4_f32(false, a, false, b2, (short)0, acc2, false, false);
            acc3 = __builtin_amdgcn_wmma_f32_16x16x4_f32(false, a, false, b3, (short)0, acc3, false, false);
            a = an; b0 = b0n; b1 = b1n; b2 = b2n; b3 = b3n;
        }
        acc0 = __builtin_amdgcn_wmma_f32_16x16x4_f32(false, a, false, b0, (short)0, acc0, false, false);
        acc1 = __builtin_amdgcn_wmma_f32_16x16x4_f32(false, a, false, b1, (short)0, acc1, false, false);
        acc2 = __builtin_amdgcn_wmma_f32_16x16x4_f32(false, a, false, b2, (short)0, acc2, false, false);
        acc3 = __builtin_amdgcn_wmma_f32_16x16x4_f32(false, a, false, b3, (short)0, acc3, false, false);
    } else {
        const float* Bk = B + (long long)(half * 2) * ldbl + col0 + l16;
        v2f a = *(const v2f*)Ap;
        v2f b0, b1, b2, b3;
        b0.x = Bk[0];  b0.y = Bk[ldbl];
        b1.x = Bk[16]; b1.y = Bk[ldbl + 16];
        b2.x = Bk[32]; b2.y = Bk[ldbl + 32];
        b3.x = Bk[48]; b3.y = Bk[ldbl + 48];
        for (int k0 = 4; k0 < K; k0 += 4) {
            Ap += 4; Bk += 4 * ldbl;
            v2f an = *(const v2f*)Ap;
            v2f b0n, b1n, b2n, b3n;
            b0n.x = Bk[0];  b0n.y = Bk[ldbl];
            b1n.x = Bk[16]; b1n.y = Bk[ldbl + 16];
            b2n.x = Bk[32]; b2n.y = Bk[ldbl + 32];
            b3n.x = Bk[48]; b3n.y = Bk[ldbl + 48];
            acc0 = __builtin_amdgcn_wmma_f32_16x16x4_f32(false, a, false, b0, (short)0, acc0, false, false);
            acc1 = __builtin_amdgcn_wmma_f32_16x16x4_f32(false, a, false, b1, (short)0, acc1, false, false);
            acc2 = __builtin_amdgcn_wmma_f32_16x16x4_f32(false, a, false, b2, (short)0, acc2, false, false);
            acc3 = __builtin_amdgcn_wmma_f32_16x16x4_f32(false, a, false, b3, (short)0, acc3, false, false);
            a = an; b0 = b0n; b1 = b1n; b2 = b2n; b3 = b3n;
        }
        acc0 = __builtin_amdgcn_wmma_f32_16x16x4_f32(false, a, false, b0, (short)0, acc0, false, false);
        acc1 = __builtin_amdgcn_wmma_f32_16x16x4_f32(false, a, false, b1, (short)0, acc1, false, false);
        acc2 = __builtin_amdgcn_wmma_f32_16x16x4_f32(false, a, false, b2, (short)0, acc2, false, false);
        acc3 = __builtin_amdgcn_wmma_f32_16x16x4_f32(false, a, false, b3, (short)0, acc3, false, false);
    }

    float* Crow = C + (long long)(row0 + half * 8) * ldc + col0 + l16;
#pragma unroll
    for (int i = 0; i < 8; ++i) {
        Crow[0]  = alpha * acc0[i];
        Crow[16] = alpha * acc1[i];
        Crow[32] = alpha * acc2[i];
        Crow[48] = alpha * acc3[i];
        Crow += ldc;
    }
}

// ---------------------------------------------------------------------------
// LayerNorm over last dim (512), biased variance, eps=1e-5. One block per row.
// ---------------------------------------------------------------------------
__global__ __launch_bounds__(256)
void layernorm_kernel(const float* __restrict__ x, const float* __restrict__ w,
                      const float* __restrict__ bb, float* __restrict__ y)
{
    __shared__ float red[256];
    const int row = blockIdx.x;
    const int t = threadIdx.x;
    const float* xr = x + (long long)row * DIMC;
    float v0 = xr[t], v1 = xr[t + 256];
    red[t] = v0 + v1;
    __syncthreads();
    for (int o = 128; o > 0; o >>= 1) {
        if (t < o) red[t] += red[t + o];
        __syncthreads();
    }
    const float mu = red[0] * (1.0f / DIMC);
    __syncthreads();
    const float d0 = v0 - mu, d1 = v1 - mu;
    red[t] = d0 * d0 + d1 * d1;
    __syncthreads();
    for (int o = 128; o > 0; o >>= 1) {
        if (t < o) red[t] += red[t + o];
        __syncthreads();
    }
    const float rstd = rsqrtf(red[0] * (1.0f / DIMC) + 1e-5f);
    float* yr = y + (long long)row * DIMC;
    yr[t]       = d0 * rstd * w[t]       + bb[t];
    yr[t + 256] = d1 * rstd * w[t + 256] + bb[t + 256];
}

// qkv [B,N,3*512] -> q,k,v [B,H,N,64]; q scaled by DHEAD^-0.5
__global__ void qkv_scatter_kernel(const float* __restrict__ qkv,
                                   float* __restrict__ q, float* __restrict__ k,
                                   float* __restrict__ v)
{
    long long idx = (long long)blockIdx.x * blockDim.x + threadIdx.x;
    int d = idx & 63; long long r = idx >> 6;
    int n = r & 2047; r >>= 11;
    int h = (int)(r & 7); int b = (int)(r >> 3);
    long long src = ((long long)(b * NSEQ + n)) * (3 * DIMC) + h * DHEAD + d;
    q[idx] = qkv[src] * QSCALE;
    k[idx] = qkv[src + DIMC];
    v[idx] = qkv[src + 2 * DIMC];
}

// mean over 8 consecutive tokens: [BH,N,64] -> [BH,M,64]
__global__ void landmark_kernel(const float* __restrict__ t, float* __restrict__ tl)
{
    long long idx = (long long)blockIdx.x * blockDim.x + threadIdx.x;
    int d = idx & 63; long long r = idx >> 6;
    int m = (int)(r & 255); int bh = (int)(r >> 8);
    const float* p = t + ((long long)bh * NSEQ + m * 8) * DHEAD + d;
    float s = 0.f;
#pragma unroll
    for (int j = 0; j < 8; ++j) s += p[j * DHEAD];
    tl[idx] = s * 0.125f;
}

// in-place row softmax, row length L. One block per row.
__global__ __launch_bounds__(256)
void softmax_kernel(float* __restrict__ a, int L)
{
    __shared__ float red[256];
    float* row = a + (long long)blockIdx.x * L;
    const int t = threadIdx.x;
    float m = -3.4e38f;
    for (int i = t; i < L; i += 256) m = fmaxf(m, row[i]);
    red[t] = m; __syncthreads();
    for (int o = 128; o > 0; o >>= 1) {
        if (t < o) red[t] = fmaxf(red[t], red[t + o]);
        __syncthreads();
    }
    m = red[0]; __syncthreads();
    float s = 0.f;
    for (int i = t; i < L; i += 256) {
        float e = __expf(row[i] - m);
        row[i] = e; s += e;
    }
    red[t] = s; __syncthreads();
    for (int o = 128; o > 0; o >>= 1) {
        if (t < o) red[t] += red[t + o];
        __syncthreads();
    }
    const float inv = 1.f / red[0];
    for (int i = t; i < L; i += 256) row[i] *= inv;
}

// |a2| summed over last dim -> [BH*M]
__global__ void abs_rowsum_kernel(const float* __restrict__ a, float* __restrict__ out)
{
    int r = blockIdx.x * blockDim.x + threadIdx.x;
    const float* p = a + (long long)r * MLAND;
    float s = 0.f;
    for (int j = 0; j < MLAND; ++j) s += fabsf(p[j]);
    out[r] = s;
}
// |a2| summed over dim -2 -> [BH*M]
__global__ void abs_colsum_kernel(const float* __restrict__ a, float* __restrict__ out)
{
    int c = blockIdx.x * blockDim.x + threadIdx.x;
    int bh = c >> 8, j = c & 255;
    const float* p = a + (long long)bh * MLAND * MLAND + j;
    float s = 0.f;
    for (int i = 0; i < MLAND; ++i) s += fabsf(p[(long long)i * MLAND]);
    out[c] = s;
}
__global__ __launch_bounds__(256)
void reduce_max_kernel(const float* __restrict__ in, int n, float* __restrict__ out)
{
    __shared__ float red[256];
    const int t = threadIdx.x;
    float m = -3.4e38f;
    for (int i = t; i < n; i += 256) m = fmaxf(m, in[i]);
    red[t] = m; __syncthreads();
    for (int o = 128; o > 0; o >>= 1) {
        if (t < o) red[t] = fmaxf(red[t], red[t + o]);
        __syncthreads();
    }
    if (t == 0) out[0] = red[0];
}
// z0 = a2^T / (rowmax * colmax)
__global__ void pinv_init_kernel(const float* __restrict__ a, const float* __restrict__ rmax,
                                 const float* __restrict__ cmax, float* __restrict__ z)
{
    long long idx = (long long)blockIdx.x * blockDim.x + threadIdx.x;
    int j = (int)(idx & 255); long long r = idx >> 8;
    int i = (int)(r & 255);   int bh = (int)(r >> 8);
    const float inv = 1.0f / (rmax[0] * cmax[0]);
    z[idx] = a[((long long)bh * MLAND + j) * MLAND + i] * inv;
}
// t = alpha*I - xz   (batched 256x256)
__global__ void id_axpy_kernel(const float* __restrict__ xz, float alpha,
                               float* __restrict__ t)
{
    long long idx = (long long)blockIdx.x * blockDim.x + threadIdx.x;
    int j = (int)(idx & 255);
    int i = (int)((idx >> 8) & 255);
    float v = -xz[idx];
    if (i == j) v += alpha;
    t[idx] = v;
}

// depthwise conv over seq dim, kernel 33, zero pad 16; v,res: [BH,N,64], w: [H,33]
__global__ __launch_bounds__(256)
void dwconv_kernel(const float* __restrict__ v, const float* __restrict__ w,
                   float* __restrict__ res)
{
    long long idx = (long long)blockIdx.x * blockDim.x + threadIdx.x;
    int d = (int)(idx & 63); long long r = idx >> 6;
    int n = (int)(r & 2047); int bh = (int)(r >> 11); int h = bh & 7;
    const float* wv = w + h * KCONV;
    const float* base = v + ((long long)bh * NSEQ) * DHEAD + d;
    float s = 0.f;
#pragma unroll
    for (int t = 0; t < KCONV; ++t) {
        int pos = n + t - KCONV / 2;
        if (pos >= 0 && pos < NSEQ) s += wv[t] * base[(long long)pos * DHEAD];
    }
    res[idx] = s;
}

// [B,H,N,64]+res -> [B,N,512]
__global__ void combine_kernel(const float* __restrict__ outh, const float* __restrict__ res,
                               float* __restrict__ outc)
{
    long long idx = (long long)blockIdx.x * blockDim.x + threadIdx.x;
    int c = (int)(idx & 511); long long r = idx >> 9;
    int n = (int)(r & 2047);  int b = (int)(r >> 11);
    int h = c >> 6, d = c & 63;
    long long src = (((long long)(b * HEADS + h) * NSEQ) + n) * DHEAD + d;
    outc[idx] = outh[src] + res[src];
}

// y = x + proj + b_out
__global__ void final_add_kernel(const float* __restrict__ x, const float* __restrict__ proj,
                                 const float* __restrict__ bout, float* __restrict__ y)
{
    long long idx = (long long)blockIdx.x * blockDim.x + threadIdx.x;
    int c = (int)(idx & 511);
    y[idx] = x[idx] + proj[idx] + bout[c];
}

// ---------------------------------------------------------------------------
static inline void launch_gemm(const float* A, const float* B, float* C,
                               int M, int N, int K, int lda, int ldb, int ldc,
                               long long sA, long long sB, long long sC,
                               int batches, int bT, float alpha, hipStream_t s)
{
    dim3 grid((N + 63) / 64, (M + 63) / 64, batches);
    if (bT)
        gemm_f32_wmma<1><<<grid, dim3(128), 0, s>>>(A, B, C, M, N, K, lda, ldb, ldc,
                                                    sA, sB, sC, alpha);
    else
        gemm_f32_wmma<0><<<grid, dim3(128), 0, s>>>(A, B, C, M, N, K, lda, ldb, ldc,
                                                    sA, sB, sC, alpha);
}

extern "C" void kernel_launch(void* const* d_in, const int* in_sizes, int n_in,
                              void* d_out, int out_size, void* d_ws, size_t ws_size,
                              hipStream_t stream)
{
    const float* x     = (const float*)d_in[0];   // [2,2048,512]
    const float* ln_w  = (const float*)d_in[1];
    const float* ln_b  = (const float*)d_in[2];
    const float* w_qkv = (const float*)d_in[3];   // [512,1536]
    const float* w_out = (const float*)d_in[4];   // [512,512]
    const float* b_out = (const float*)d_in[5];
    const float* w_res = (const float*)d_in[6];   // [8,1,33,1]

    float* out_y    = (float*)d_out;                                  // [2,2048,512]
    float* out_attn = (float*)d_out + (long long)ROWS * DIMC;         // [2,8,2048,2048]

    // ---- workspace layout (floats) ----
    float* W = (float*)d_ws;
    float* xn   = W;  W += (long long)ROWS * DIMC;          // 4096x512
    float* qkv  = W;  W += (long long)ROWS * 3 * DIMC;      // 4096x1536
    float* q    = W;  W += (long long)BH * NSEQ * DHEAD;
    float* k    = W;  W += (long long)BH * NSEQ * DHEAD;
    float* v    = W;  W += (long long)BH * NSEQ * DHEAD;
    float* q_l  = W;  W += (long long)BH * MLAND * DHEAD;
    float* k_l  = W;  W += (long long)BH * MLAND * DHEAD;
    float* a1   = W;  W += (long long)BH * NSEQ * MLAND;
    float* a2   = W;  W += (long long)BH * MLAND * MLAND;
    float* a3   = W;  W += (long long)BH * MLAND * NSEQ;
    float* zA   = W;  W += (long long)BH * MLAND * MLAND;
    float* zB   = W;  W += (long long)BH * MLAND * MLAND;
    float* xz   = W;  W += (long long)BH * MLAND * MLAND;
    float* tb   = W;  W += (long long)BH * MLAND * MLAND;
    float* ub   = W;  W += (long long)BH * MLAND * MLAND;
    float* B1   = W;  W += (long long)BH * NSEQ * MLAND;
    float* a3v  = W;  W += (long long)BH * MLAND * DHEAD;
    float* outh = W;  W += (long long)BH * NSEQ * DHEAD;
    float* resb = W;  W += (long long)BH * NSEQ * DHEAD;
    float* outc = W;  W += (long long)ROWS * DIMC;
    float* proj = W;  W += (long long)ROWS * DIMC;
    float* rsum = W;  W += BH * MLAND;
    float* csum = W;  W += BH * MLAND;
    float* rmax = W;  W += 1;
    float* cmax = W;  W += 1;

    const long long sQ  = (long long)NSEQ * DHEAD;      // per-bh stride of q/k/v
    const long long sL  = (long long)MLAND * DHEAD;     // per-bh stride of q_l/k_l
    const long long sA1 = (long long)NSEQ * MLAND;
    const long long sM2 = (long long)MLAND * MLAND;
    const long long sA3 = (long long)MLAND * NSEQ;

    // 1) LayerNorm
    layernorm_kernel<<<ROWS, 256, 0, stream>>>(x, ln_w, ln_b, xn);

    // 2) QKV projection: [4096,512] @ [512,1536]
    launch_gemm(xn, w_qkv, qkv, ROWS, 3 * DIMC, DIMC, DIMC, 3 * DIMC, 3 * DIMC,
                0, 0, 0, 1, 0, 1.0f, stream);

    // 3) split into q,k,v heads (scale q)
    qkv_scatter_kernel<<<(BH * NSEQ * DHEAD) / 256, 256, 0, stream>>>(qkv, q, k, v);

    // 4) landmark means
    landmark_kernel<<<(BH * MLAND * DHEAD) / 256, 256, 0, stream>>>(q, q_l);
    landmark_kernel<<<(BH * MLAND * DHEAD) / 256, 256, 0, stream>>>(k, k_l);

    // 5) sim1 = q @ k_l^T, sim2 = q_l @ k_l^T, sim3 = q_l @ k^T
    launch_gemm(q,   k_l, a1, NSEQ,  MLAND, DHEAD, DHEAD, DHEAD, MLAND, sQ, sL, sA1, BH, 1, 1.0f, stream);
    launch_gemm(q_l, k_l, a2, MLAND, MLAND, DHEAD, DHEAD, DHEAD, MLAND, sL, sL, sM2, BH, 1, 1.0f, stream);
    launch_gemm(q_l, k,   a3, MLAND, NSEQ,  DHEAD, DHEAD, DHEAD, NSEQ,  sL, sQ, sA3, BH, 1, 1.0f, stream);

    // 6) softmaxes (in place)
    softmax_kernel<<<BH * NSEQ,  256, 0, stream>>>(a1, MLAND);
    softmax_kernel<<<BH * MLAND, 256, 0, stream>>>(a2, MLAND);
    softmax_kernel<<<BH * MLAND, 256, 0, stream>>>(a3, NSEQ);

    // 7) Moore-Penrose init: z = a2^T / (max(rowsum)*max(colsum))
    abs_rowsum_kernel<<<BH * MLAND / 256, 256, 0, stream>>>(a2, rsum);
    abs_colsum_kernel<<<BH * MLAND / 256, 256, 0, stream>>>(a2, csum);
    reduce_max_kernel<<<1, 256, 0, stream>>>(rsum, BH * MLAND, rmax);
    reduce_max_kernel<<<1, 256, 0, stream>>>(csum, BH * MLAND, cmax);
    pinv_init_kernel<<<(int)(BH * sM2 / 256), 256, 0, stream>>>(a2, rmax, cmax, zA);

    // 8) Newton-Schulz iterations: z = 0.25 z (13I - xz(15I - xz(7I - xz)))
    float* zp = zA; float* zq = zB;
    const int nPinvElems = (int)(BH * sM2 / 256);
    for (int it = 0; it < PINV_ITERS; ++it) {
        launch_gemm(a2, zp, xz, MLAND, MLAND, MLAND, MLAND, MLAND, MLAND, sM2, sM2, sM2, BH, 0, 1.0f, stream);
        id_axpy_kernel<<<nPinvElems, 256, 0, stream>>>(xz,  7.0f, tb);
        launch_gemm(xz, tb, ub, MLAND, MLAND, MLAND, MLAND, MLAND, MLAND, sM2, sM2, sM2, BH, 0, 1.0f, stream);
        id_axpy_kernel<<<nPinvElems, 256, 0, stream>>>(ub, 15.0f, tb);
        launch_gemm(xz, tb, ub, MLAND, MLAND, MLAND, MLAND, MLAND, MLAND, sM2, sM2, sM2, BH, 0, 1.0f, stream);
        id_axpy_kernel<<<nPinvElems, 256, 0, stream>>>(ub, 13.0f, tb);
        launch_gemm(zp, tb, zq, MLAND, MLAND, MLAND, MLAND, MLAND, MLAND, sM2, sM2, sM2, BH, 0, 0.25f, stream);
        float* tmp = zp; zp = zq; zq = tmp;
    }

    // 9) B1 = a1 @ a2_inv   [2048x256]
    launch_gemm(a1, zp, B1, NSEQ, MLAND, MLAND, MLAND, MLAND, MLAND, sA1, sM2, sA1, BH, 0, 1.0f, stream);

    // 10) a3v = a3 @ v  [256x64];  outh = B1 @ a3v  [2048x64]
    launch_gemm(a3, v,   a3v,  MLAND, DHEAD, NSEQ,  NSEQ,  DHEAD, DHEAD, sA3, sQ, sL, BH, 0, 1.0f, stream);
    launch_gemm(B1, a3v, outh, NSEQ,  DHEAD, MLAND, MLAND, DHEAD, DHEAD, sA1, sL, sQ, BH, 0, 1.0f, stream);

    // 11) attn output = B1 @ a3  [2048x2048] -> second region of d_out (HBM-bound step)
    launch_gemm(B1, a3, out_attn, NSEQ, NSEQ, MLAND, MLAND, NSEQ, NSEQ,
                sA1, sA3, (long long)NSEQ * NSEQ, BH, 0, 1.0f, stream);

    // 12) depthwise conv residual + head merge
    dwconv_kernel<<<(BH * NSEQ * DHEAD) / 256, 256, 0, stream>>>(v, w_res, resb);
    combine_kernel<<<(ROWS * DIMC) / 256, 256, 0, stream>>>(outh, resb, outc);

    // 13) output projection + residual: y = x + outc @ w_out + b_out
    launch_gemm(outc, w_out, proj, ROWS, DIMC, DIMC, DIMC, DIMC, DIMC, 0, 0, 0, 1, 0, 1.0f, stream);
    final_add_kernel<<<(ROWS * DIMC) / 256, 256, 0, stream>>>(x, proj, b_out, out_y);
}